// NIBLayer_28106265985425
// MI455X (gfx1250) — compile-verified
//
#include <hip/hip_runtime.h>
#include <math.h>
#include <stdint.h>

static constexpr int kN = 8192;
static constexpr int kD = 128;
static constexpr int kTILE = 128;
static constexpr int kSTRIDE = 132;  // 128 + 4 pad -> banks 4m+k, conflict-free for 16-lane groups
static constexpr int kNT = kN / kTILE;               // 64 tile-rows
static constexpr int kNPAIRS = kNT * (kNT + 1) / 2;  // 2080 upper-triangular tile pairs

typedef __attribute__((ext_vector_type(2))) float v2f;
typedef __attribute__((ext_vector_type(8))) float v8f;

__device__ __forceinline__ float softplus_dev(const float* __restrict__ phi) {
    return log1pf(__expf(phi[0]));
}

// ---------------------------------------------------------------------------
// Kernel 1: out = x + var*noise ; sq[i] = sum_d x[i,d]^2 ; rowS[i] = 0
// ---------------------------------------------------------------------------
__global__ __launch_bounds__(128) void prep_kernel(const float* __restrict__ x,
                                                   const float* __restrict__ phi,
                                                   const float* __restrict__ noise,
                                                   float* __restrict__ out,
                                                   float* __restrict__ sq,
                                                   float* __restrict__ rowS) {
    const int i = blockIdx.x;
    const int d = threadIdx.x;
    const float var = softplus_dev(phi);
    const float v = x[i * kD + d];
    out[i * kD + d] = v + var * noise[i * kD + d];

    float p = v * v;
    p += __shfl_xor(p, 16, 32);
    p += __shfl_xor(p, 8, 32);
    p += __shfl_xor(p, 4, 32);
    p += __shfl_xor(p, 2, 32);
    p += __shfl_xor(p, 1, 32);

    __shared__ float lred[4];
    const int lane = d & 31, wid = d >> 5;
    if (lane == 0) lred[wid] = p;
    __syncthreads();
    if (d == 0) {
        sq[i] = lred[0] + lred[1] + lred[2] + lred[3];
        rowS[i] = 0.0f;
    }
}

// ---------------------------------------------------------------------------
// Kernel 2: one 128x128 Gram tile per upper-triangular tile pair (bi <= bj).
// Staging: GLOBAL_LOAD_ASYNC_TO_LDS_B128 (ASYNCcnt path, no VGPR round-trip).
// Compute: 8 waves; each wave owns a 32x64 sub-tile (2x4 WMMA subtiles) ->
// 6 ds_load_b64 feed 8 v_wmma_f32_16x16x4_f32 per k-step.
// Off-diagonal tiles are harvested twice via row-sums AND column-sums.
// ---------------------------------------------------------------------------
__global__ __launch_bounds__(256) void gram_kernel(const float* __restrict__ x,
                                                   const float* __restrict__ sq,
                                                   const float* __restrict__ phi,
                                                   float* __restrict__ rowS) {
    __shared__ float As[kTILE][kSTRIDE];
    __shared__ float Bs[kTILE][kSTRIDE];

    // linear block id -> (bi, bj) with bi <= bj
    int p = blockIdx.x;
    int bi = 0, span = kNT;
    while (p >= span) { p -= span; ++bi; --span; }
    const int bj = bi + p;
    const int i0 = bi * kTILE;
    const int j0 = bj * kTILE;
    const bool diag = (bi == bj);

    const int tid = threadIdx.x;

    // Async stage: 128x128 floats per operand, 16B per lane per op,
    // direct global -> LDS (tracked by ASYNCcnt).
    #pragma unroll
    for (int q = 0; q < (kTILE * kD / 4) / 256; ++q) {  // 16 iterations
        const int idx = q * 256 + tid;
        const int r = idx >> 5;        // 32 float4 per row
        const int c4 = (idx & 31) * 4;
        const float* ga = &x[(size_t)(i0 + r) * kD + c4];
        const float* gb = &x[(size_t)(j0 + r) * kD + c4];
        const uint32_t la = (uint32_t)(uintptr_t)&As[r][c4];  // LDS byte offset
        const uint32_t lb = (uint32_t)(uintptr_t)&Bs[r][c4];
        asm volatile("global_load_async_to_lds_b128 %0, %1, off"
                     :: "v"(la), "v"(ga) : "memory");
        asm volatile("global_load_async_to_lds_b128 %0, %1, off"
                     :: "v"(lb), "v"(gb) : "memory");
    }
    // each wave waits for its own async copies, then the barrier publishes LDS
    asm volatile("s_wait_asynccnt 0x0" ::: "memory");
    __syncthreads();

    const int w = tid >> 5;        // wave 0..7
    const int lane = tid & 31;
    const int m = lane & 15;       // M (A) / N (B) index within 16
    const int hi = lane >> 4;      // half-wave: selects K pair of the frag
    const int wr = w >> 1;         // 0..3 -> rows [wr*32, wr*32+32)
    const int wc = w & 1;          // 0..1 -> cols [wc*64, wc*64+64)

    v8f acc[2][4];
    #pragma unroll
    for (int r = 0; r < 2; ++r)
        #pragma unroll
        for (int t = 0; t < 4; ++t) acc[r][t] = v8f{};

    for (int k = 0; k < kD; k += 4) {
        const int kb = k + hi * 2;  // lanes 0-15: K=k,k+1 ; lanes 16-31: K=k+2,k+3
        v2f a[2], b[4];
        #pragma unroll
        for (int r = 0; r < 2; ++r) {
            const float2 t2 = *(const float2*)&As[wr * 32 + r * 16 + m][kb];
            a[r].x = t2.x; a[r].y = t2.y;
        }
        #pragma unroll
        for (int t = 0; t < 4; ++t) {
            const float2 t2 = *(const float2*)&Bs[wc * 64 + t * 16 + m][kb];
            b[t].x = t2.x; b[t].y = t2.y;
        }
        #pragma unroll
        for (int r = 0; r < 2; ++r)
            #pragma unroll
            for (int t = 0; t < 4; ++t)
                acc[r][t] = __builtin_amdgcn_wmma_f32_16x16x4_f32(
                    false, a[r], false, b[t], (short)0, acc[r][t], false, false);
    }

    // Epilogue: e = exp(-0.5*max(si+sj-2g,0)/var); row sums always,
    // column sums (transposed contribution) for off-diagonal tiles.
    const float var = softplus_dev(phi);
    const float scale = -0.5f / var;

    float sj[4];
    #pragma unroll
    for (int t = 0; t < 4; ++t) sj[t] = sq[j0 + wc * 64 + t * 16 + m];

    float cs[4] = {0.0f, 0.0f, 0.0f, 0.0f};

    #pragma unroll
    for (int r = 0; r < 2; ++r) {
        #pragma unroll
        for (int v = 0; v < 8; ++v) {
            // C/D layout: VGPR v holds row v (lanes 0-15) or v+8 (lanes 16-31)
            const int grow = i0 + wr * 32 + r * 16 + v + hi * 8;
            const float si = sq[grow];
            float rs = 0.0f;
            #pragma unroll
            for (int t = 0; t < 4; ++t) {
                const float d2 = fmaxf(si + sj[t] - 2.0f * acc[r][t][v], 0.0f);
                const float e = __expf(scale * d2);
                rs += e;
                cs[t] += e;
            }
            // reduce across the 16 lanes (columns) of this half-wave
            rs += __shfl_xor(rs, 1, 32);
            rs += __shfl_xor(rs, 2, 32);
            rs += __shfl_xor(rs, 4, 32);
            rs += __shfl_xor(rs, 8, 32);
            if (m == 0) atomicAdd(&rowS[grow], rs);
        }
    }

    if (!diag) {
        #pragma unroll
        for (int t = 0; t < 4; ++t) {
            // merge the two half-waves (rows v and v+8 of both strips)
            const float c = cs[t] + __shfl_xor(cs[t], 16, 32);
            if (hi == 0) atomicAdd(&rowS[j0 + wc * 64 + t * 16 + m], c);
        }
    }
}

// ---------------------------------------------------------------------------
// Kernel 3: IXT = (log N - mean_i log(rowS[i])) / log 2
// ---------------------------------------------------------------------------
__global__ __launch_bounds__(256) void finalize_kernel(const float* __restrict__ rowS,
                                                       float* __restrict__ ixt) {
    __shared__ float red[256];
    float s = 0.0f;
    for (int i = threadIdx.x; i < kN; i += 256) s += __logf(rowS[i]);
    red[threadIdx.x] = s;
    __syncthreads();
    for (int stride = 128; stride > 0; stride >>= 1) {
        if (threadIdx.x < stride) red[threadIdx.x] += red[threadIdx.x + stride];
        __syncthreads();
    }
    if (threadIdx.x == 0) {
        const float kde = red[0] / (float)kN;
        ixt[0] = (logf((float)kN) - kde) / logf(2.0f);
    }
}

// ---------------------------------------------------------------------------
extern "C" void kernel_launch(void* const* d_in, const int* in_sizes, int n_in,
                              void* d_out, int out_size, void* d_ws, size_t ws_size,
                              hipStream_t stream) {
    const float* x     = (const float*)d_in[0];
    const float* phi   = (const float*)d_in[1];
    const float* noise = (const float*)d_in[2];

    float* out  = (float*)d_out;          // [N*D] out, then [1] IXT
    float* sq   = (float*)d_ws;           // N floats
    float* rowS = sq + kN;                // N floats

    prep_kernel<<<kN, kD, 0, stream>>>(x, phi, noise, out, sq, rowS);

    gram_kernel<<<kNPAIRS, 256, 0, stream>>>(x, sq, phi, rowS);

    finalize_kernel<<<1, 256, 0, stream>>>(rowS, out + (size_t)kN * kD);
}